// align_inter_aligned_23639499997224
// MI455X (gfx1250) — compile-verified
//
#include <hip/hip_runtime.h>
#include <stdint.h>

#define EPSF 1e-5f

#define WAVES_PER_BLOCK 4
#define WAVE_TILE 128                              // boxes per wave per tile
#define BLOCK_TILE (WAVES_PER_BLOCK * WAVE_TILE)   // 512 boxes per block-iteration
#define BOX_BYTES 28
#define WAVE_TILE_BYTES (WAVE_TILE * BOX_BYTES)    // 3584 bytes (7 x 512B passes)

// ---- CDNA5 async global->LDS path (gfx1250) -------------------------------
#if defined(__has_builtin)
#if __has_builtin(__builtin_amdgcn_global_load_async_to_lds_b128) && \
    __has_builtin(__builtin_amdgcn_s_wait_asynccnt)
#define HAVE_ASYNC 1
#endif
#endif
#ifndef HAVE_ASYNC
#define HAVE_ASYNC 0
#endif

typedef int v4i __attribute__((ext_vector_type(4)));
typedef __attribute__((address_space(1))) v4i* as1_v4i;
typedef __attribute__((address_space(3))) v4i* as3_v4i;

__device__ __forceinline__ as1_v4i to_as1(const void* p) {
  // global generic pointers share representation with addrspace(1)
  return (as1_v4i)(uintptr_t)p;
}
__device__ __forceinline__ as3_v4i to_as3(const void* p) {
  // low 32 bits of a generic LDS pointer are the LDS byte address (aperture rule)
  return (as3_v4i)(uint32_t)(uintptr_t)p;
}

// Stage one wave-tile (3584B of g + 3584B of q) into wave-private LDS.
// 7 + 7 per-lane B128 async transfers, fully coalesced (16B/lane, 512B/wave/pass).
__device__ __forceinline__ void stage_tile(const char* gsrc, const char* qsrc,
                                           char* gdst, char* qdst, int lane) {
#if HAVE_ASYNC
#pragma unroll
  for (int j = 0; j < 7; ++j) {
    const int off = j * 512 + lane * 16;
    __builtin_amdgcn_global_load_async_to_lds_b128(to_as1(gsrc + off), to_as3(gdst + off), 0, 0);
  }
#pragma unroll
  for (int j = 0; j < 7; ++j) {
    const int off = j * 512 + lane * 16;
    __builtin_amdgcn_global_load_async_to_lds_b128(to_as1(qsrc + off), to_as3(qdst + off), 0, 0);
  }
#else
  // synchronous fallback: regular b128 loads + ds stores
#pragma unroll
  for (int j = 0; j < 7; ++j) {
    const int off = j * 512 + lane * 16;
    *(float4*)(gdst + off) = *(const float4*)(gsrc + off);
    *(float4*)(qdst + off) = *(const float4*)(qsrc + off);
  }
#endif
}

__device__ __forceinline__ void wait_prev_tile() {
#if HAVE_ASYNC
  __builtin_amdgcn_s_wait_asynccnt(14);   // 14 newer ops (next tile) may remain in flight
#endif
  __asm__ volatile("" ::: "memory");
}
__device__ __forceinline__ void wait_all_tiles() {
#if HAVE_ASYNC
  __builtin_amdgcn_s_wait_asynccnt(0);
#endif
  __asm__ volatile("" ::: "memory");
}

// ---- per-box math ----------------------------------------------------------
// Axis interval overlap / union widths; planes xoz, xoy, yoz.
__device__ __forceinline__ void box_planes(const float* G, const float* Q, float* r6) {
  const float gxl = G[0] - 0.5f * G[3], gxh = G[0] + 0.5f * G[3];
  const float gyl = G[1] - 0.5f * G[4], gyh = G[1] + 0.5f * G[4];
  const float gzl = G[2] - 0.5f * G[5], gzh = G[2] + 0.5f * G[5];
  const float qxl = Q[0] - 0.5f * Q[3], qxh = Q[0] + 0.5f * Q[3];
  const float qyl = Q[1] - 0.5f * Q[4], qyh = Q[1] + 0.5f * Q[4];
  const float qzl = Q[2] - 0.5f * Q[5], qzh = Q[2] + 0.5f * Q[5];

  const float iwx = fminf(gxh, qxh) - fmaxf(gxl, qxl) + EPSF;
  const float iwy = fminf(gyh, qyh) - fmaxf(gyl, qyl) + EPSF;
  const float iwz = fminf(gzh, qzh) - fmaxf(gzl, qzl) + EPSF;
  const float mwx = fmaxf(gxh, qxh) - fminf(gxl, qxl) + EPSF;
  const float mwy = fmaxf(gyh, qyh) - fminf(gyl, qyl) + EPSF;
  const float mwz = fmaxf(gzh, qzh) - fminf(gzl, qzl) + EPSF;

  r6[0] = (iwx > 0.f && iwz > 0.f) ? iwx * iwz : 0.f;  // inter_xoz
  r6[1] = mwx * mwz;                                   // mbr_xoz
  r6[2] = (iwx > 0.f && iwy > 0.f) ? iwx * iwy : 0.f;  // inter_xoy
  r6[3] = mwx * mwy;                                   // mbr_xoy
  r6[4] = (iwy > 0.f && iwz > 0.f) ? iwy * iwz : 0.f;  // inter_yoz
  r6[5] = mwy * mwz;                                   // mbr_yoz
}

// ---- main tiled kernel: async double-buffered LDS staging ------------------
__global__ __launch_bounds__(WAVES_PER_BLOCK * 32) void planes_lds_kernel(
    const float* __restrict__ g, const float* __restrict__ q,
    float* __restrict__ out, int nTiles, int N) {
  __shared__ alignas(16) char smem[WAVES_PER_BLOCK][2][2][WAVE_TILE_BYTES];  // 56KB
  const int lane = threadIdx.x & 31;
  const int wave = threadIdx.x >> 5;

  int t = (int)blockIdx.x;
  if (t >= nTiles) return;
  const int stride = (int)gridDim.x;
  int buf = 0;

  {  // prologue: stage first tile into buffer 0
    const size_t base = (size_t)t * BLOCK_TILE + (size_t)wave * WAVE_TILE;
    stage_tile((const char*)(g + base * 7), (const char*)(q + base * 7),
               smem[wave][0][0], smem[wave][0][1], lane);
  }

  for (; t < nTiles; t += stride) {
    const int tn = t + stride;
    if (tn < nTiles) {
      const size_t baseN = (size_t)tn * BLOCK_TILE + (size_t)wave * WAVE_TILE;
      stage_tile((const char*)(g + baseN * 7), (const char*)(q + baseN * 7),
                 smem[wave][buf ^ 1][0], smem[wave][buf ^ 1][1], lane);
      wait_prev_tile();   // current buffer's 14 transfers complete; next 14 in flight
    } else {
      wait_all_tiles();
    }

    const char* gb = smem[wave][buf][0];
    const char* qb = smem[wave][buf][1];

    // each lane owns 4 consecutive boxes: 28 floats = 7 aligned float4 LDS reads
    float ga[28], qa[28];
#pragma unroll
    for (int j = 0; j < 7; ++j) {
      const float4 v = *(const float4*)(gb + lane * 112 + j * 16);
      ga[4 * j + 0] = v.x; ga[4 * j + 1] = v.y; ga[4 * j + 2] = v.z; ga[4 * j + 3] = v.w;
      const float4 u = *(const float4*)(qb + lane * 112 + j * 16);
      qa[4 * j + 0] = u.x; qa[4 * j + 1] = u.y; qa[4 * j + 2] = u.z; qa[4 * j + 3] = u.w;
    }

    float r[4][6];
#pragma unroll
    for (int b = 0; b < 4; ++b) box_planes(&ga[7 * b], &qa[7 * b], r[b]);

    // coalesced float4 stores to all 6 output streams
    const size_t idx = (size_t)t * BLOCK_TILE + (size_t)wave * WAVE_TILE + (size_t)(4 * lane);
#pragma unroll
    for (int k = 0; k < 6; ++k) {
      float4 v;
      v.x = r[0][k]; v.y = r[1][k]; v.z = r[2][k]; v.w = r[3][k];
      *(float4*)(out + (size_t)k * (size_t)N + idx) = v;
    }
    buf ^= 1;
  }
}

// ---- scalar tail / generic fallback ----------------------------------------
__global__ void planes_direct_kernel(const float* __restrict__ g,
                                     const float* __restrict__ q,
                                     float* __restrict__ out, int N, int start) {
  const int i = start + (int)(blockIdx.x * blockDim.x + threadIdx.x);
  if (i >= N) return;
  float G[7], Q[7];
#pragma unroll
  for (int c = 0; c < 7; ++c) {
    G[c] = g[(size_t)i * 7 + c];
    Q[c] = q[(size_t)i * 7 + c];
  }
  float r[6];
  box_planes(G, Q, r);
#pragma unroll
  for (int k = 0; k < 6; ++k) out[(size_t)k * (size_t)N + i] = r[k];
}

extern "C" void kernel_launch(void* const* d_in, const int* in_sizes, int n_in,
                              void* d_out, int out_size, void* d_ws, size_t ws_size,
                              hipStream_t stream) {
  const float* g = (const float*)d_in[0];
  const float* q = (const float*)d_in[1];
  float* out = (float*)d_out;
  const int N = in_sizes[0] / 7;
  if (N <= 0) return;

  int nTiles = 0, start = 0;
  if ((N % 4) == 0) {  // tiled kernel's float4 output stores need N % 4 == 0
    nTiles = N / BLOCK_TILE;
    start = nTiles * BLOCK_TILE;
  }
  if (nTiles > 0) {
    const int grid = nTiles < 1024 ? nTiles : 1024;
    planes_lds_kernel<<<grid, WAVES_PER_BLOCK * 32, 0, stream>>>(g, q, out, nTiles, N);
  }
  const int rem = N - start;
  if (rem > 0) {
    planes_direct_kernel<<<(rem + 255) / 256, 256, 0, stream>>>(g, q, out, N, start);
  }
}